// SemanticBank_18476949307683
// MI455X (gfx1250) — compile-verified
//
#include <hip/hip_runtime.h>
#include <math.h>

// Problem constants from the reference
#define N_ROWS 16384
#define C_DIM  128
#define K_CLS  8192
#define ALPHA  0.85f

#define NCHUNK 4                     // split of K for parallelism
#define WAVES_PER_BLOCK 8
#define TILES_PER_CHUNK (K_CLS / 16 / NCHUNK)   // 128 column tiles per chunk
#define NEG_INF (-3.402823466e38f)

typedef __attribute__((ext_vector_type(2))) float v2f;
typedef __attribute__((ext_vector_type(8))) float v8f;
typedef __attribute__((ext_vector_type(4))) int   v4i;

#if __has_builtin(__builtin_amdgcn_global_load_async_to_lds_b128)
#define HAVE_ASYNC_LDS 1
#else
#define HAVE_ASYNC_LDS 0
#endif

#if HAVE_ASYNC_LDS
__device__ __forceinline__ void sb_wait_async_le2() {
#if __has_builtin(__builtin_amdgcn_s_wait_asynccnt)
    __builtin_amdgcn_s_wait_asynccnt(2);
#else
    asm volatile("s_wait_asynccnt 0x2" ::: "memory");
#endif
}
__device__ __forceinline__ void sb_wait_async_le0() {
#if __has_builtin(__builtin_amdgcn_s_wait_asynccnt)
    __builtin_amdgcn_s_wait_asynccnt(0);
#else
    asm volatile("s_wait_asynccnt 0x0" ::: "memory");
#endif
}
#endif

// ---------------- kernel 1: winner[k] = -1 ----------------
__global__ void sb_init_winner(int* __restrict__ winner) {
    int k = blockIdx.x * blockDim.x + threadIdx.x;
    if (k < K_CLS) winner[k] = -1;
}

// ---------------- kernel 2: last-write-wins vote ----------------
__global__ void sb_vote(const int* __restrict__ label, int* __restrict__ winner) {
    int i = blockIdx.x * blockDim.x + threadIdx.x;
    if (i < N_ROWS) atomicMax(&winner[label[i]], i);
}

// ---------------- kernel 3: new_bank = bank (*alpha + f[winner]) ----------------
__global__ void sb_update_bank(const float* __restrict__ bank,
                               const float* __restrict__ f,
                               const int* __restrict__ winner,
                               float* __restrict__ nb) {
    int idx = blockIdx.x * blockDim.x + threadIdx.x;   // K*C total
    int k = idx >> 7;          // / C_DIM
    int c = idx & (C_DIM - 1);
    int w = winner[k];
    float v = bank[idx];
    if (w >= 0) v = v * ALPHA + f[(size_t)w * C_DIM + c];
    nb[idx] = v;
}

// ---------------- kernel 4: fused GEMM (WMMA f32 16x16x4) + online logsumexp ----
// grid: (N/16/8, NCHUNK), block: 256 (8 waves). Wave w owns row tile
// blockIdx.x*8+w; all waves in a block walk the same column tiles and share the
// 16x128 bank tile through double-buffered LDS filled by async global->LDS DMA.
__global__ void sb_gemm_lse(const float* __restrict__ f,
                            const float* __restrict__ nb,
                            const int*   __restrict__ label,
                            float* __restrict__ rowmax,   // [NCHUNK][N]
                            float* __restrict__ rowsum,   // [NCHUNK][N]
                            float* __restrict__ pos_buf)  // [N]
{
    __shared__ float bsh[2][16][132];   // +4 pad: conflict-free ds_load_b64 reads

    const int tid  = threadIdx.x;
    const int wave = tid >> 5;
    const int lane = tid & 31;
    const int rlow = lane & 15;
    const int half = lane >> 4;

    const int rowBase  = (blockIdx.x * WAVES_PER_BLOCK + wave) * 16;
    const int chunk    = blockIdx.y;
    const int colBase0 = chunk * (TILES_PER_CHUNK * 16);

    // A strip (16 rows x 128) resident in VGPRs: 32 float2 fragments per lane.
    // ISA layout: lane holds row (lane&15), elements kbase+2*(lane>>4), +1.
    v2f a_frag[32];
    const float* arow = f + (size_t)(rowBase + rlow) * C_DIM;
    #pragma unroll
    for (int ks = 0; ks < 32; ++ks)
        a_frag[ks] = *(const v2f*)(arow + ks * 4 + 2 * half);

    // Labels for the 8 rows this lane accumulates (C/D layout: row r + 8*half)
    int lbl[8];
    #pragma unroll
    for (int r = 0; r < 8; ++r) lbl[r] = label[rowBase + half * 8 + r];

    float m[8], s[8];
    #pragma unroll
    for (int r = 0; r < 8; ++r) { m[r] = NEG_INF; s[r] = 0.f; }

#if HAVE_ASYNC_LDS
    // Issue a 16x128 f32 tile as async global->LDS DMA: 512 lane-transfers of
    // 16B; 2 instructions per thread (256 threads). Per-lane LDS address VGPR
    // lets us keep the padded (132-float pitch) layout.
    auto issue_tile = [&](int tileIdx, int buf) {
        const int colB = colBase0 + tileIdx * 16;
        #pragma unroll
        for (int q = 0; q < 2; ++q) {
            int flat4 = q * 256 + tid;        // which 4-float group (0..511)
            int col   = flat4 >> 5;           // 32 groups per 128-float row
            int kk    = (flat4 & 31) * 4;
            const float* src = nb + (size_t)(colB + col) * C_DIM + kk;
            __builtin_amdgcn_global_load_async_to_lds_b128(
                (__attribute__((address_space(1))) v4i*)src,
                (__attribute__((address_space(3))) v4i*)&bsh[buf][col][kk],
                0, 0);
        }
    };
    issue_tile(0, 0);
#endif

    for (int tn = 0; tn < TILES_PER_CHUNK; ++tn) {
        const int colBase = colBase0 + tn * 16;
#if HAVE_ASYNC_LDS
        const int cur = tn & 1;
        if (tn + 1 < TILES_PER_CHUNK) {
            // buf cur^1 was last read in the previous iteration; end-of-iter barrier
            // below guarantees all waves are done with it.
            issue_tile(tn + 1, cur ^ 1);
            sb_wait_async_le2();   // async loads retire in order: current tile done
        } else {
            sb_wait_async_le0();
        }
        __syncthreads();           // all waves' DMA for buf[cur] visible
        const float (*bt)[132] = bsh[cur];
#else
        __syncthreads();           // previous iteration's LDS reads complete
        {   // cooperative stage of 16x128 bank tile: 8 floats per thread
            int flat = tid * 8;
            int col  = flat >> 7;        // / 128
            int kk   = flat & 127;
            const float* src = nb + (size_t)(colBase + col) * C_DIM + kk;
            float4 p0 = *(const float4*)(src);
            float4 p1 = *(const float4*)(src + 4);
            *(float4*)&bsh[0][col][kk]     = p0;
            *(float4*)&bsh[0][col][kk + 4] = p1;
        }
        __syncthreads();
        const float (*bt)[132] = bsh[0];
#endif

        // 16x16 logits tile: 32 x V_WMMA_F32_16X16X4_F32 over the C=128 reduction
        v8f acc = {};
        #pragma unroll
        for (int ks = 0; ks < 32; ++ks) {
            v2f b = *(const v2f*)&bt[rlow][ks * 4 + 2 * half];
            acc = __builtin_amdgcn_wmma_f32_16x16x4_f32(
                      /*neg_a=*/false, a_frag[ks],
                      /*neg_b=*/false, b,
                      /*c_mod=*/(short)0, acc,
                      /*reuse_a=*/false, /*reuse_b=*/false);
        }

        // online logsumexp update + positive-logit capture
        const int colc = colBase + rlow;
        #pragma unroll
        for (int r = 0; r < 8; ++r) {
            float x = acc[r];                       // row rowBase+8*half+r, col colc
            if (lbl[r] == colc) pos_buf[rowBase + half * 8 + r] = x;
            float mn = fmaxf(m[r], x);
            s[r] = s[r] * __expf(m[r] - mn) + __expf(x - mn);
            m[r] = mn;
        }

#if HAVE_ASYNC_LDS
        __syncthreads();   // all reads of buf[cur] done before it is re-filled
#endif
    }

    // reduce (m,s) across the 16 lanes of each half-wave (rows stay per-half)
    #pragma unroll
    for (int r = 0; r < 8; ++r) {
        float mr = m[r], sr = s[r];
        #pragma unroll
        for (int off = 1; off < 16; off <<= 1) {
            float mo = __shfl_xor(mr, off, 32);
            float so = __shfl_xor(sr, off, 32);
            float mn = fmaxf(mr, mo);
            sr = sr * __expf(mr - mn) + so * __expf(mo - mn);
            mr = mn;
        }
        if (rlow == 0) {
            int row = rowBase + half * 8 + r;
            rowmax[(size_t)chunk * N_ROWS + row] = mr;
            rowsum[(size_t)chunk * N_ROWS + row] = sr;
        }
    }
}

// ---------------- kernel 5: per-row lse - pos, block partial sums ----------------
__global__ void sb_finalize_rows(const float* __restrict__ rowmax,
                                 const float* __restrict__ rowsum,
                                 const float* __restrict__ pos_buf,
                                 float* __restrict__ partial) {
    __shared__ float red[256];
    int i = blockIdx.x * blockDim.x + threadIdx.x;

    float M = NEG_INF;
    #pragma unroll
    for (int c = 0; c < NCHUNK; ++c) M = fmaxf(M, rowmax[(size_t)c * N_ROWS + i]);
    float S = 0.f;
    #pragma unroll
    for (int c = 0; c < NCHUNK; ++c)
        S += rowsum[(size_t)c * N_ROWS + i] * __expf(rowmax[(size_t)c * N_ROWS + i] - M);

    float v = M + logf(S) - pos_buf[i];

    red[threadIdx.x] = v;
    __syncthreads();
    for (int off = 128; off > 0; off >>= 1) {
        if (threadIdx.x < off) red[threadIdx.x] += red[threadIdx.x + off];
        __syncthreads();
    }
    if (threadIdx.x == 0) partial[blockIdx.x] = red[0];
}

// ---------------- kernel 6: deterministic final sum ----------------
__global__ void sb_final_sum(const float* __restrict__ partial, float* __restrict__ out,
                             int nparts) {
    if (threadIdx.x == 0) {
        float t = 0.f;
        for (int i = 0; i < nparts; ++i) t += partial[i];   // fixed order
        out[0] = t / (float)N_ROWS;
    }
}

extern "C" void kernel_launch(void* const* d_in, const int* in_sizes, int n_in,
                              void* d_out, int out_size, void* d_ws, size_t ws_size,
                              hipStream_t stream) {
    const float* f     = (const float*)d_in[0];   // [N, C]
    const int*   label = (const int*)  d_in[1];   // [N]
    const float* bank  = (const float*)d_in[2];   // [K, C]
    float* out = (float*)d_out;

    // workspace carve-up
    char* ws = (char*)d_ws;
    float* nb      = (float*)ws;                         ws += (size_t)K_CLS * C_DIM * 4;
    int*   winner  = (int*)ws;                           ws += (size_t)K_CLS * 4;
    float* rowmax  = (float*)ws;                         ws += (size_t)NCHUNK * N_ROWS * 4;
    float* rowsum  = (float*)ws;                         ws += (size_t)NCHUNK * N_ROWS * 4;
    float* posb    = (float*)ws;                         ws += (size_t)N_ROWS * 4;
    float* partial = (float*)ws;

    sb_init_winner<<<K_CLS / 256, 256, 0, stream>>>(winner);
    sb_vote<<<N_ROWS / 256, 256, 0, stream>>>(label, winner);
    sb_update_bank<<<(K_CLS * C_DIM) / 256, 256, 0, stream>>>(bank, f, winner, nb);

    dim3 grid(N_ROWS / 16 / WAVES_PER_BLOCK, NCHUNK);   // (128, 4)
    sb_gemm_lse<<<grid, 256, 0, stream>>>(f, nb, label, rowmax, rowsum, posb);

    sb_finalize_rows<<<N_ROWS / 256, 256, 0, stream>>>(rowmax, rowsum, posb, partial);
    sb_final_sum<<<1, 32, 0, stream>>>(partial, out, N_ROWS / 256);
}